// HardCrossEntropy2d_32229434589493
// MI455X (gfx1250) — compile-verified
//
#include <hip/hip_runtime.h>
#include <hip/hip_bf16.h>
#include <math.h>

// -----------------------------------------------------------------------------
// OHEM cross-entropy on MI455X (gfx1250, wave32).
// Memory-bound: 318MB logits streamed once (~14us @ 23.3TB/s) + ~64MB aux
// traffic (L2-resident: 192MB L2 holds the 16MB nll scratch).
// Exact k-th largest selection via 2-level float-bit histogram (prob>=0 =>
// IEEE bits are monotone). Pass-1 histogram uses per-block LDS aggregation
// (64KB of the 320KB/WGP LDS) to kill same-address L2 atomic serialization.
// Final reduction uses V_WMMA_F32_16X16X4_F32 with a ones B-matrix as an
// exact f32 row-sum reducer (CDNA5 matrix path, codegen-confirmed).
// -----------------------------------------------------------------------------

#define NB 8
#define NC 19
#define NH 512
#define NW 1024
#define PLANE (NH * NW)          // 524288 = 2^19
#define NP (NB * PLANE)          // 4194304 = 2^22
#define H1BINS 16384
#define H2BINS 65536

typedef __attribute__((ext_vector_type(2))) float v2f;
typedef __attribute__((ext_vector_type(8))) float v8f;

// ---------------- pass 1: per-pixel NLL + level-1 histogram ------------------
__global__ void __launch_bounds__(1024)
k_pass1(const float* __restrict__ pred,
        const int*   __restrict__ tgt,
        float*       __restrict__ nllArr,
        unsigned*    __restrict__ hist1) {
  __shared__ unsigned lhist[H1BINS];           // 64KB LDS per block
  for (int i = threadIdx.x; i < H1BINS; i += blockDim.x) lhist[i] = 0u;
  __syncthreads();

  const int stride = gridDim.x * blockDim.x;
  for (int p = blockIdx.x * blockDim.x + threadIdx.x; p < NP; p += stride) {
    const int n   = p >> 19;            // / PLANE
    const int rem = p & (PLANE - 1);
    const float* base = pred + (((size_t)n * NC) << 19) + rem;

    const int label = tgt[p];
    const bool valid = (label != 255);
    const int lab = valid ? label : 0;

    float x[NC];
    float m = -INFINITY;
#pragma unroll
    for (int c = 0; c < NC; ++c) {
      x[c] = base[(size_t)c << 19];     // coalesced per channel plane
      m = fmaxf(m, x[c]);
    }
    float s = 0.0f;
#pragma unroll
    for (int c = 0; c < NC; ++c) s += __expf(x[c] - m);

    float xt = x[0];
#pragma unroll
    for (int c = 1; c < NC; ++c) if (c == lab) xt = x[c];

    const float logp = xt - m - __logf(s);
    const float nll  = -logp;

    nllArr[p] = valid ? nll : __uint_as_float(0x7FC00000u); // NaN = ignored

    if (valid) {
      const float prob = __expf(-nll);                  // same recipe everywhere
      const unsigned key = __float_as_uint(prob) >> 16; // prob>=0 -> monotone
      atomicAdd(&lhist[key], 1u);                       // LDS atomic (hidden)
    }
  }

  __syncthreads();
  for (int i = threadIdx.x; i < H1BINS; i += blockDim.x) {
    const unsigned c = lhist[i];
    if (c) atomicAdd(&hist1[i], c);                     // one flush per touched bin
  }
}

// -------- level-1 select: bin containing rank r, and rank-within-bin ---------
__global__ void k_select1(const unsigned* __restrict__ hist1,
                          unsigned*       __restrict__ sel) {
  __shared__ unsigned csum[256];
  __shared__ unsigned suf[256];
  __shared__ int rsh;
  const int t = threadIdx.x;

  unsigned s = 0;
#pragma unroll 4
  for (int i = 0; i < H1BINS / 256; ++i) s += hist1[t * (H1BINS / 256) + i];
  csum[t] = s;
  __syncthreads();

  if (t == 0) {
    unsigned run = 0;
    for (int i = 255; i >= 0; --i) { suf[i] = run; run += csum[i]; }
    const unsigned nv = run;                         // num_valid = sum of hist
    const unsigned nk = (unsigned)(int)((float)nv * 0.25f);
    rsh = (nk > 0u) ? (int)nk - 1 : 0;               // rank from the top
  }
  __syncthreads();

  const int r = rsh;
  unsigned cum = suf[t];                             // count strictly above chunk
  for (int i = H1BINS / 256 - 1; i >= 0; --i) {
    const int bin = t * (H1BINS / 256) + i;
    const unsigned h = hist1[bin];
    if (h > 0u && (int)cum <= r && r < (int)(cum + h)) {
      sel[0] = (unsigned)bin;            // level-1 bin (top 16 prob bits)
      sel[1] = (unsigned)(r - (int)cum); // rank within bin, from the top
    }
    cum += h;
  }
}

// ------------- refine pass: level-2 histogram of candidate bin ---------------
__global__ void k_refine(const float*    __restrict__ nllArr,
                         const unsigned* __restrict__ sel,
                         unsigned*       __restrict__ hist2) {
  const unsigned b1 = sel[0];
  const int stride = gridDim.x * blockDim.x;
  for (int p = blockIdx.x * blockDim.x + threadIdx.x; p < NP; p += stride) {
    const float v = nllArr[p];
    if (v == v) {                                    // valid (non-NaN)
      const unsigned bits = __float_as_uint(__expf(-v));
      if ((bits >> 16) == b1) atomicAdd(&hist2[bits & 0xFFFFu], 1u);
    }
  }
}

// ---------------- level-2 select: exact threshold bit pattern ----------------
__global__ void k_select2(const unsigned* __restrict__ hist2,
                          unsigned*       __restrict__ sel) {
  __shared__ unsigned csum[256];
  __shared__ unsigned suf[256];
  __shared__ int r2sh;
  __shared__ unsigned b1sh;
  const int t = threadIdx.x;

  unsigned s = 0;
  for (int i = 0; i < H2BINS / 256; ++i) s += hist2[t * (H2BINS / 256) + i];
  csum[t] = s;
  __syncthreads();

  if (t == 0) {
    unsigned run = 0;
    for (int i = 255; i >= 0; --i) { suf[i] = run; run += csum[i]; }
    r2sh = (int)sel[1];
    b1sh = sel[0];
  }
  __syncthreads();

  const int r2 = r2sh;
  unsigned cum = suf[t];
  for (int i = H2BINS / 256 - 1; i >= 0; --i) {
    const int bin = t * (H2BINS / 256) + i;
    const unsigned h = hist2[bin];
    if (h > 0u && (int)cum <= r2 && r2 < (int)(cum + h)) {
      sel[2] = (b1sh << 16) | (unsigned)bin;         // threshold float bits
    }
    cum += h;
  }
}

// ------------- loss pass: kept sum / count, WMMA f32 block reduce ------------
__global__ void k_loss(const float*    __restrict__ nllArr,
                       const unsigned* __restrict__ sel,
                       float*          __restrict__ acc) {   // acc[0]=sum acc[1]=cnt
  const float thr = __uint_as_float(sel[2]);
  float s = 0.0f, k = 0.0f;

  const int stride = gridDim.x * blockDim.x;
  for (int p = blockIdx.x * blockDim.x + threadIdx.x; p < NP; p += stride) {
    const float v = nllArr[p];
    if (v == v) {
      const float prob = __expf(-v);
      if (prob <= thr) { s += v; k += 1.0f; }
    }
  }

  __shared__ float ls[256];
  __shared__ float lk[256];
  const int t = threadIdx.x;
  ls[t] = s; lk[t] = k;
  __syncthreads();
  if (t < 128) { ls[t] += ls[t + 128]; lk[t] += lk[t + 128]; }
  __syncthreads();
  if (t < 64)  { ls[t] += ls[t + 64];  lk[t] += lk[t + 64];  }
  __syncthreads();

  if (t < 32) {
    // Exact f32 reduction of 64 partials via one WMMA per quantity:
    // A (16x4) holds the 64 values, B = ones(4x16) => D rows are row-sums.
    // A VGPR layout (ISA 7.12.2): lane l<16 holds A[l][0..1], lane 16+l holds
    // A[l][2..3]. Map value i -> A[i/4][i%4].
    const int base = (t & 15) * 4 + (t >> 4) * 2;
    v2f as; as.x = ls[base]; as.y = ls[base + 1];
    v2f ak; ak.x = lk[base]; ak.y = lk[base + 1];
    v2f bo; bo.x = 1.0f; bo.y = 1.0f;
    v8f c0 = {};
    v8f ds = __builtin_amdgcn_wmma_f32_16x16x4_f32(
        false, as, false, bo, (short)0, c0, false, false);
    v8f dk = __builtin_amdgcn_wmma_f32_16x16x4_f32(
        false, ak, false, bo, (short)0, c0, false, false);
    float rs = ds[0] + ds[1] + ds[2] + ds[3] + ds[4] + ds[5] + ds[6] + ds[7];
    float rk = dk[0] + dk[1] + dk[2] + dk[3] + dk[4] + dk[5] + dk[6] + dk[7];
    // lane l has rows 0..7 (l<16) or rows 8..15 (l>=16) of column l%16 summed.
    rs += __shfl_xor(rs, 16, 32);
    rk += __shfl_xor(rk, 16, 32);
    if (t == 0) {
      atomicAdd(&acc[0], rs);
      atomicAdd(&acc[1], rk);
    }
  }
}

__global__ void k_final(const float* __restrict__ acc, float* __restrict__ out) {
  out[0] = acc[0] / fmaxf(acc[1], 1.0f);
}

// -----------------------------------------------------------------------------
extern "C" void kernel_launch(void* const* d_in, const int* in_sizes, int n_in,
                              void* d_out, int out_size, void* d_ws, size_t ws_size,
                              hipStream_t stream) {
  (void)in_sizes; (void)n_in; (void)out_size; (void)ws_size;

  const float* pred = (const float*)d_in[0];
  const int*   tgt  = (const int*)d_in[1];
  float*       out  = (float*)d_out;

  float*    nllArr = (float*)d_ws;                              // NP floats
  unsigned* hist1  = (unsigned*)((char*)d_ws + (size_t)NP * 4); // 16384
  unsigned* hist2  = hist1 + H1BINS;                            // 65536
  unsigned* sel    = hist2 + H2BINS;                            // sel[0..7]
  float*    acc    = (float*)(sel + 4);                         // sum, cnt

  // zero histograms + selection/accumulator state (graph-capturable)
  hipMemsetAsync(hist1, 0, (size_t)(H1BINS + H2BINS + 8) * 4, stream);

  // 256 blocks x 1024 threads: 32 waves/block, 64KB LDS/block (<=5 blocks/WGP),
  // 16K pixels per block so LDS-hist flush amortizes to <=1 atomic/bin/block.
  k_pass1  <<<256, 1024, 0, stream>>>(pred, tgt, nllArr, hist1);
  k_select1<<<1,    256, 0, stream>>>(hist1, sel);
  k_refine <<<2048, 256, 0, stream>>>(nllArr, sel, hist2);
  k_select2<<<1,    256, 0, stream>>>(hist2, sel);
  k_loss   <<<2048, 256, 0, stream>>>(nllArr, sel, acc);
  k_final  <<<1,      1, 0, stream>>>(acc, out);
}